// PhysicsInformedLayer_9019431321744
// MI455X (gfx1250) — compile-verified
//
#include <hip/hip_runtime.h>
#include <hip/hip_bf16.h>

typedef float v2f __attribute__((ext_vector_type(2)));
typedef float v8f __attribute__((ext_vector_type(8)));

#define V_MIN 0.95f
#define V_MAX 1.05f
#define PBW   10.0f

// ---------------------------------------------------------------------------
// Zero the scatter workspace (p_calc, q_calc, acc) each call.
// ---------------------------------------------------------------------------
__global__ void pf_zero_kernel(float* __restrict__ ws, int n) {
    int i = blockIdx.x * blockDim.x + threadIdx.x;
    if (i < n) ws[i] = 0.0f;
}

// ---------------------------------------------------------------------------
// Edge kernel: one thread per edge, inner loop over batches.
//  - from/to indices loaded once per edge (amortized over B batches)
//  - r/x loads are fully coalesced streams (the HBM-bound part)
//  - v_mag/v_ang gathers hit the L2-resident 1.6MB arrays
//  - scatter via native fp32 global atomics (relaxed, agent scope)
// ---------------------------------------------------------------------------
__global__ void pf_edge_kernel(const float* __restrict__ v_mag,
                               const float* __restrict__ v_ang,
                               const float* __restrict__ r_line,
                               const float* __restrict__ x_line,
                               const int*   __restrict__ from_idx,
                               const int*   __restrict__ to_idx,
                               float* __restrict__ p_calc,
                               float* __restrict__ q_calc,
                               int B, int N, int E) {
    int e = blockIdx.x * blockDim.x + threadIdx.x;
    if (e >= E) return;

    const int fi = from_idx[e];
    const int ti = to_idx[e];

    for (int b = 0; b < B; ++b) {
        const size_t be = (size_t)b * (size_t)E + (size_t)e;
        const size_t bn = (size_t)b * (size_t)N;

        if (b + 1 < B) {  // prefetch next batch's r/x rows (3.2MB stride)
            __builtin_prefetch(&r_line[be + (size_t)E], 0, 0);
            __builtin_prefetch(&x_line[be + (size_t)E], 0, 0);
        }

        const float r  = r_line[be];
        const float x  = x_line[be];
        const float z2 = r * r + x * x;
        const float iz = 1.0f / z2;
        const float g  =  r * iz;
        const float bl = -x * iz;

        const float vi = v_mag[bn + fi];
        const float vj = v_mag[bn + ti];
        const float th = v_ang[bn + fi] - v_ang[bn + ti];

        float st, ct;
        __sincosf(th, &st, &ct);

        const float vv = vi * vj;
        const float p  = vv * (g * ct + bl * st);
        const float q  = vv * (g * st - bl * ct);

        __hip_atomic_fetch_add(&p_calc[bn + fi], p,
                               __ATOMIC_RELAXED, __HIP_MEMORY_SCOPE_AGENT);
        __hip_atomic_fetch_add(&q_calc[bn + fi], q,
                               __ATOMIC_RELAXED, __HIP_MEMORY_SCOPE_AGENT);
    }
}

// ---------------------------------------------------------------------------
// Bus kernel: mismatch + voltage penalty, passthrough outputs, and a
// wave-level reduction done on the matrix pipe:
//   D = A x ones  (V_WMMA_F32_16X16X4_F32)  =>  D[m][n] = rowsum(A,m)
// Each lane packs its local partial into A (a[0]=partial, a[1]=0), so after
// the WMMA, lane L<16 summing its 8 C VGPRs holds the total of lanes
// {0..7, 16..23}, lane L>=16 holds {8..15, 24..31}. Lanes 0 and 16 atomically
// add -> exact wave sum, 2 atomics per wave. EXEC is all-ones here (grid-
// stride loop has reconverged), satisfying the WMMA EXEC restriction.
// ---------------------------------------------------------------------------
__global__ void pf_bus_kernel(const float* __restrict__ v_mag,
                              const float* __restrict__ v_ang,
                              const float* __restrict__ p_inj,
                              const float* __restrict__ q_inj,
                              const float* __restrict__ p_calc,
                              const float* __restrict__ q_calc,
                              float* __restrict__ out_vmag,
                              float* __restrict__ out_vang,
                              float* __restrict__ acc,
                              int total) {
    float local = 0.0f;
    const int stride = gridDim.x * blockDim.x;
    for (int i = blockIdx.x * blockDim.x + threadIdx.x; i < total; i += stride) {
        const float vm = v_mag[i];
        out_vmag[i] = vm;          // passthrough output 0
        out_vang[i] = v_ang[i];    // passthrough output 1

        const float pm = p_calc[i] - p_inj[i];
        const float qm = q_calc[i] - q_inj[i];
        const float lo = fmaxf(V_MIN - vm, 0.0f);
        const float hi = fmaxf(vm - V_MAX, 0.0f);
        local += PBW * (pm * pm + qm * qm) + lo * lo + hi * hi;
    }

    // Wave reduction via V_WMMA_F32_16X16X4_F32 with B = ones.
    v2f a;    a[0] = local; a[1] = 0.0f;
    v2f ones; ones[0] = 1.0f; ones[1] = 1.0f;
    v8f c = {};
    c = __builtin_amdgcn_wmma_f32_16x16x4_f32(
            /*neg_a=*/false, a, /*neg_b=*/false, ones,
            /*c_mod=*/(short)0, c, /*reuse_a=*/false, /*reuse_b=*/false);

    const float csum = ((c[0] + c[1]) + (c[2] + c[3])) +
                       ((c[4] + c[5]) + (c[6] + c[7]));
    const int lane = threadIdx.x & 31;
    if (lane == 0 || lane == 16) {
        __hip_atomic_fetch_add(acc, csum,
                               __ATOMIC_RELAXED, __HIP_MEMORY_SCOPE_AGENT);
    }
}

// ---------------------------------------------------------------------------
// Finalize: loss = acc / B  (means over batch = totals / B; the 10x power
// weight was folded into the per-bus contribution).
// ---------------------------------------------------------------------------
__global__ void pf_finalize_kernel(const float* __restrict__ acc,
                                   float* __restrict__ out_loss,
                                   float invB) {
    if (blockIdx.x == 0 && threadIdx.x == 0) {
        out_loss[0] = acc[0] * invB;
    }
}

extern "C" void kernel_launch(void* const* d_in, const int* in_sizes, int n_in,
                              void* d_out, int out_size, void* d_ws, size_t ws_size,
                              hipStream_t stream) {
    const float* v_mag  = (const float*)d_in[0];
    const float* v_ang  = (const float*)d_in[1];
    const float* r_line = (const float*)d_in[2];
    const float* x_line = (const float*)d_in[3];
    const float* p_inj  = (const float*)d_in[4];
    const float* q_inj  = (const float*)d_in[5];
    const int*   eidx   = (const int*)d_in[6];

    const int BN = in_sizes[0];       // B*N
    const int E  = in_sizes[6] / 2;   // edge_index is (2, E)
    const int BE = in_sizes[2];       // B*E
    const int B  = BE / E;
    const int N  = BN / B;

    float* p_calc = (float*)d_ws;        // BN floats
    float* q_calc = p_calc + BN;         // BN floats
    float* acc    = q_calc + BN;         // 1 float

    float* out_vmag = (float*)d_out;
    float* out_vang = out_vmag + BN;
    float* out_loss = out_vang + BN;

    // 1) zero scatter workspace + accumulator
    const int nz = 2 * BN + 1;
    pf_zero_kernel<<<(nz + 255) / 256, 256, 0, stream>>>(p_calc, nz);

    // 2) edge scatter (thread per edge, loop over batches)
    pf_edge_kernel<<<(E + 255) / 256, 256, 0, stream>>>(
        v_mag, v_ang, r_line, x_line, eidx, eidx + E,
        p_calc, q_calc, B, N, E);

    // 3) bus mismatch + loss reduction + passthrough outputs
    int blocks = (BN + 255) / 256;
    if (blocks > 1024) blocks = 1024;
    pf_bus_kernel<<<blocks, 256, 0, stream>>>(
        v_mag, v_ang, p_inj, q_inj, p_calc, q_calc,
        out_vmag, out_vang, acc, BN);

    // 4) finalize scalar loss
    pf_finalize_kernel<<<1, 32, 0, stream>>>(acc, out_loss, 1.0f / (float)B);
}